// Head_21629455302923
// MI455X (gfx1250) — compile-verified
//
#include <hip/hip_runtime.h>
#include <hip/hip_bf16.h>

#define N_NODES   200000
#define N_EDGES   3200000
#define NUM_GRAPHS 512
#define CHANNELS  32
#define GH        64
#define CONVS     8
#define FEATS     288   // (CONVS+1)*CHANNELS
#define EPSV      1e-5f

typedef __attribute__((ext_vector_type(2))) float v2f;
typedef __attribute__((ext_vector_type(8))) float v8f;

// float atomic max via signed/unsigned int ordering trick (valid with -inf init)
__device__ __forceinline__ void atomicMaxF(float* addr, float v) {
  if (v >= 0.0f) atomicMax((int*)addr, __float_as_int(v));
  else           atomicMin((unsigned int*)addr, __float_as_uint(v));
}

__global__ void init_kernel(float* pooled, float* counts) {
  int i = blockIdx.x * blockDim.x + threadIdx.x;
  if (i < NUM_GRAPHS * FEATS) pooled[i] = -INFINITY;
  if (i < NUM_GRAPHS) counts[i] = 0.0f;
}

__global__ void embed_kernel(const int* __restrict__ x, const int* __restrict__ batch,
                             const float* __restrict__ emb,
                             float* __restrict__ h, float* __restrict__ aggr,
                             float* __restrict__ counts, float* __restrict__ pooled) {
  int i = blockIdx.x * blockDim.x + threadIdx.x;
  if (i >= N_NODES * CHANNELS) return;
  int node = i >> 5, c = i & 31;
  float v = emb[x[node] * CHANNELS + c];
  h[i] = v;
  aggr[i] = 0.0f;
  int b = batch[node];
  atomicMaxF(&pooled[b * FEATS + c], v);     // xs[0] slice of segment_max
  if (c == 0) atomicAdd(&counts[b], 1.0f);
}

// one wave32 per edge, lane = channel: 128B coalesced gather + coalesced atomic scatter
__global__ void edge_kernel(const int* __restrict__ src, const int* __restrict__ dst,
                            const float* __restrict__ h, float* __restrict__ aggr) {
  int t = blockIdx.x * blockDim.x + threadIdx.x;
  int e = t >> 5;
  int lane = t & 31;
  if (e >= N_EDGES) return;
  int s = src[e], d = dst[e];
  atomicAdd(&aggr[d * CHANNELS + lane], h[s * CHANNELS + lane]);
}

// Node MLP with fp32 WMMA: relu((h+aggr)@Wh + bh)@Wo + bo -> z ; per-graph sum/sumsq
__global__ __launch_bounds__(256) void mlp_kernel(
    const float* __restrict__ h, const float* __restrict__ aggr,
    const float* __restrict__ Wh, const float* __restrict__ bh,
    const float* __restrict__ Wo, const float* __restrict__ bo,
    const int* __restrict__ batch,
    float* __restrict__ zbuf, float* __restrict__ gsum, float* __restrict__ gsumsq) {
  __shared__ float sWhT[64 * 34];       // WhT[j][c] = Wh[c][j], stride 34 (8B aligned, bank-safe)
  __shared__ float sWoT[32 * 66];       // WoT[j][c] = Wo[c][j], stride 66
  __shared__ float sStage[8][16 * 66];  // per-wave: zin (cols 0..31) then relu(z1) (cols 0..63)

  const int tid  = threadIdx.x;
  const int wave = tid >> 5;
  const int lane = tid & 31;
  const int half = lane >> 4;           // K-half selector for 32-bit A/B WMMA layouts
  const int lm   = lane & 15;           // M (or N) index within a 16-wide tile

  // load + transpose weights into LDS (2048 elements each)
  for (int idx = tid; idx < 2048; idx += 256) {
    int c = idx >> 6, j = idx & 63;     // Wh[32][64]
    sWhT[j * 34 + c] = Wh[c * 64 + j];
    int c2 = idx >> 5, j2 = idx & 31;   // Wo[64][32]
    sWoT[j2 * 66 + c2] = Wo[c2 * 32 + j2];
  }
  __syncthreads();

  const int base = blockIdx.x * 128 + wave * 16;

  // stage zin = h + aggr (coalesced rows; clamp tail, no divergence before WMMA)
  for (int r = 0; r < 16; ++r) {
    int node = base + r; if (node >= N_NODES) node = N_NODES - 1;
    sStage[wave][r * 66 + lane] = h[node * 32 + lane] + aggr[node * 32 + lane];
  }

  // GEMM1: [16x32] @ Wh[32x64], K in steps of 4, 4 N-tiles
  v8f acc1[4] = {};
#pragma unroll
  for (int k = 0; k < 8; ++k) {
    int kk = 4 * k + 2 * half;
    v2f a = *(const v2f*)&sStage[wave][lm * 66 + kk];      // A: 16x4 f32 layout
#pragma unroll
    for (int t = 0; t < 4; ++t) {
      v2f b = *(const v2f*)&sWhT[(16 * t + lm) * 34 + kk]; // B: 4x16 f32 layout
      acc1[t] = __builtin_amdgcn_wmma_f32_16x16x4_f32(false, a, false, b,
                                                      (short)0, acc1[t], false, false);
    }
  }

  // bias + relu, restage as A-matrix source (LDS ops in a wave are in-order)
#pragma unroll
  for (int t = 0; t < 4; ++t) {
    float bias = bh[16 * t + lm];
#pragma unroll
    for (int r = 0; r < 8; ++r) {
      float v = acc1[t][r] + bias;
      sStage[wave][(r + 8 * half) * 66 + 16 * t + lm] = fmaxf(v, 0.0f);
    }
  }

  // GEMM2: [16x64] @ Wo[64x32], 16 K-steps, 2 N-tiles
  v8f acc2[2] = {};
#pragma unroll
  for (int k = 0; k < 16; ++k) {
    int kk = 4 * k + 2 * half;
    v2f a = *(const v2f*)&sStage[wave][lm * 66 + kk];
#pragma unroll
    for (int t = 0; t < 2; ++t) {
      v2f b = *(const v2f*)&sWoT[(16 * t + lm) * 66 + kk];
      acc2[t] = __builtin_amdgcn_wmma_f32_16x16x4_f32(false, a, false, b,
                                                      (short)0, acc2[t], false, false);
    }
  }

  // epilogue: z = acc2 + bo ; store z, accumulate per-graph sum / sumsq
#pragma unroll
  for (int t = 0; t < 2; ++t) {
    int col = 16 * t + lm;
    float bias = bo[col];
#pragma unroll
    for (int r = 0; r < 8; ++r) {
      int node = base + r + 8 * half;
      if (node < N_NODES) {
        float v = acc2[t][r] + bias;
        zbuf[node * 32 + col] = v;
        int b = batch[node];
        atomicAdd(&gsum[b * 32 + col], v);
        atomicAdd(&gsumsq[b * 32 + col], v * v);
      }
    }
  }
}

// instance-norm + residual relu; also zeroes aggr for the next conv, updates pooled max
__global__ void norm_kernel(const float* __restrict__ zbuf, const float* __restrict__ gsum,
                            const float* __restrict__ gsumsq, const float* __restrict__ counts,
                            const int* __restrict__ batch,
                            float* __restrict__ h, float* __restrict__ aggr,
                            float* __restrict__ pooled, int colOff) {
  int i = blockIdx.x * blockDim.x + threadIdx.x;
  if (i >= N_NODES * CHANNELS) return;
  int node = i >> 5, c = i & 31;
  int b = batch[node];
  float cnt  = counts[b];
  float mean = gsum[b * 32 + c] / cnt;
  float var  = gsumsq[b * 32 + c] / cnt - mean * mean;
  float zn   = (zbuf[i] - mean) * rsqrtf(var + EPSV);
  float hn   = fmaxf(h[i] + zn, 0.0f);
  h[i] = hn;
  aggr[i] = 0.0f;
  atomicMaxF(&pooled[b * FEATS + colOff + c], hn);
}

// head: relu(pooled @ W_hid + b_hid) @ W_out + b_out   (one block of 64 per graph)
__global__ void head_kernel(const float* __restrict__ pooled,
                            const float* __restrict__ Whid, const float* __restrict__ bhid,
                            const float* __restrict__ Wout, const float* __restrict__ bout,
                            float* __restrict__ out) {
  __shared__ float sp[FEATS];
  __shared__ float sh[GH];
  int g = blockIdx.x, t = threadIdx.x;
  for (int k = t; k < FEATS; k += 64) sp[k] = pooled[g * FEATS + k];
  __syncthreads();
  float acc = bhid[t];
  for (int k = 0; k < FEATS; ++k) acc += sp[k] * Whid[k * GH + t];
  sh[t] = fmaxf(acc, 0.0f);
  __syncthreads();
  if (t < 32) {
    float o = bout[t];
    for (int k = 0; k < GH; ++k) o += sh[k] * Wout[k * 32 + t];
    out[g * 32 + t] = o;
  }
}

extern "C" void kernel_launch(void* const* d_in, const int* in_sizes, int n_in,
                              void* d_out, int out_size, void* d_ws, size_t ws_size,
                              hipStream_t stream) {
  const int*   x     = (const int*)d_in[0];
  const int*   eidx  = (const int*)d_in[1];
  const int*   batch = (const int*)d_in[2];
  const float* emb   = (const float*)d_in[3];
  const float* Wh    = (const float*)d_in[4];
  const float* bh    = (const float*)d_in[5];
  const float* Wo    = (const float*)d_in[6];
  const float* bo    = (const float*)d_in[7];
  const float* W_hid = (const float*)d_in[8];
  const float* b_hid = (const float*)d_in[9];
  const float* W_out = (const float*)d_in[10];
  const float* b_out = (const float*)d_in[11];
  float* out = (float*)d_out;

  float* ws     = (float*)d_ws;
  float* h      = ws;
  float* aggr   = h    + (size_t)N_NODES * CHANNELS;
  float* zbuf   = aggr + (size_t)N_NODES * CHANNELS;
  float* gsum   = zbuf + (size_t)N_NODES * CHANNELS;
  float* gsumsq = gsum   + NUM_GRAPHS * CHANNELS;
  float* counts = gsumsq + NUM_GRAPHS * CHANNELS;
  float* pooled = counts + NUM_GRAPHS;

  const int* src = eidx;
  const int* dst = eidx + N_EDGES;

  {
    int n = NUM_GRAPHS * FEATS;
    init_kernel<<<(n + 255) / 256, 256, 0, stream>>>(pooled, counts);
  }
  {
    int n = N_NODES * CHANNELS;
    embed_kernel<<<(n + 255) / 256, 256, 0, stream>>>(x, batch, emb, h, aggr, counts, pooled);
  }
  for (int i = 0; i < CONVS; ++i) {
    hipMemsetAsync(gsum, 0, (size_t)2 * NUM_GRAPHS * CHANNELS * sizeof(float), stream);
    {
      int n = N_EDGES * 32;  // 102.4M threads, fits in int
      edge_kernel<<<(n + 255) / 256, 256, 0, stream>>>(src, dst, h, aggr);
    }
    {
      int blocks = (N_NODES + 127) / 128;
      mlp_kernel<<<blocks, 256, 0, stream>>>(h, aggr,
                                             Wh + (size_t)i * CHANNELS * GH, bh + (size_t)i * GH,
                                             Wo + (size_t)i * GH * CHANNELS, bo + (size_t)i * CHANNELS,
                                             batch, zbuf, gsum, gsumsq);
    }
    {
      int n = N_NODES * CHANNELS;
      norm_kernel<<<(n + 255) / 256, 256, 0, stream>>>(zbuf, gsum, gsumsq, counts, batch,
                                                       h, aggr, pooled, (i + 1) * CHANNELS);
    }
  }
  head_kernel<<<NUM_GRAPHS, 64, 0, stream>>>(pooled, W_hid, b_hid, W_out, b_out, out);
}